// GCN_47931835023574
// MI455X (gfx1250) — compile-verified
//
#include <hip/hip_runtime.h>

typedef __attribute__((ext_vector_type(2))) float v2f;
typedef __attribute__((ext_vector_type(8))) float v8f;

#define DIM 128
#define XS_STRIDE 132   // pad 16x128 LDS tile to kill 512B-stride bank conflicts

// ---------------- utility kernels ----------------

__global__ void k_zero(float* __restrict__ p, int n) {
    int i = blockIdx.x * blockDim.x + threadIdx.x;
    if (i < n) p[i] = 0.0f;
}

__global__ void k_deg_count(const int* __restrict__ col, float* __restrict__ deg, int e) {
    int i = blockIdx.x * blockDim.x + threadIdx.x;
    if (i < e) atomicAdd(&deg[col[i]], 1.0f);
}

__global__ void k_dinv(const float* __restrict__ deg, float* __restrict__ dinv, int n) {
    int i = blockIdx.x * blockDim.x + threadIdx.x;
    if (i < n) {
        float d = deg[i];
        dinv[i] = (d > 0.0f) ? rsqrtf(d) : 0.0f;
    }
}

// out[n][d] = bias[d]  (bias folded into the scatter accumulator init)
__global__ void k_bias_init(float* __restrict__ out, const float* __restrict__ bias, long total) {
    long i = (long)blockIdx.x * blockDim.x + threadIdx.x;
    if (i < total) out[i] = bias[i & (DIM - 1)];
}

__global__ void k_relu(float* __restrict__ p, long total) {
    long i = (long)blockIdx.x * blockDim.x + threadIdx.x;
    if (i < total) p[i] = fmaxf(p[i], 0.0f);
}

// ---------------- fp32 WMMA GEMM: H = X @ W  (X: n x 128, W: 128 x 128) ----------------
// Block = 256 threads (8 waves). Each block handles a 16-row M tile; wave w owns
// output columns [16w, 16w+16). K loop: 32 x v_wmma_f32_16x16x4_f32.
__global__ void __launch_bounds__(256)
k_gemm_wmma(const float* __restrict__ X, const float* __restrict__ Wm,
            float* __restrict__ H, int n) {
    __shared__ float xs[16 * XS_STRIDE];

    const int m0  = blockIdx.x * 16;
    const int tid = threadIdx.x;

    // Stage the 16x128 A tile into LDS (coalesced: 8 floats per thread).
    {
        int r = tid >> 4;          // 0..15
        int c = (tid & 15) * 8;    // 0..120
        const float* src = X + (size_t)(m0 + r) * DIM + c;
        float4 v0, v1;
        if (m0 + r < n) {
            v0 = *(const float4*)(src);
            v1 = *(const float4*)(src + 4);
        } else {
            v0 = make_float4(0.f, 0.f, 0.f, 0.f);
            v1 = v0;
        }
        *(float4*)&xs[r * XS_STRIDE + c]     = v0;
        *(float4*)&xs[r * XS_STRIDE + c + 4] = v1;
    }
    __syncthreads();

    const int wave = tid >> 5;
    const int lane = tid & 31;
    const int lo   = lane & 15;   // M index (A) / N index (B, C/D)
    const int hi   = lane >> 4;   // selects K pair / M+8 half
    const int col  = wave * 16 + lo;

    v8f acc = {};
    #pragma unroll 4
    for (int k = 0; k < DIM; k += 4) {
        // A fragment: A[M=lo][K = k + 2*hi + v]  (v = VGPR 0/1)
        v2f a = *(const v2f*)&xs[lo * XS_STRIDE + k + 2 * hi];
        // B fragment: B[K = k + 2*hi + v][N = col]  (W row-major 128x128, L2/L0 resident)
        v2f b;
        b.x = Wm[(k + 2 * hi) * DIM + col];
        b.y = Wm[(k + 2 * hi + 1) * DIM + col];
        acc = __builtin_amdgcn_wmma_f32_16x16x4_f32(
            /*neg_a=*/false, a, /*neg_b=*/false, b,
            /*c_mod=*/(short)0, acc, /*reuse_a=*/false, /*reuse_b=*/false);
    }

    // D[M = v + 8*hi][N = col].  Hot path: full tile (always true for n % 16 == 0)
    // -> 8 plain stores with immediate offsets, no per-element exec juggling.
    float* hbase = H + (size_t)(m0 + 8 * hi) * DIM + col;
    if (m0 + 16 <= n) {           // uniform per-block branch
        #pragma unroll
        for (int v = 0; v < 8; ++v) {
            hbase[(size_t)v * DIM] = acc[v];
        }
    } else {
        #pragma unroll
        for (int v = 0; v < 8; ++v) {
            if (m0 + v + 8 * hi < n) hbase[(size_t)v * DIM] = acc[v];
        }
    }
}

// ---------------- edge scatter: out[col] += norm * h[row]  ----------------
// One wave (32 lanes) per edge; each lane moves float4 (512B row per wave).
__global__ void __launch_bounds__(256)
k_scatter(const float* __restrict__ H, const int* __restrict__ row,
          const int* __restrict__ col, const float* __restrict__ dinv,
          float* __restrict__ out, int e) {
    long gtid = (long)blockIdx.x * blockDim.x + threadIdx.x;
    int  edge = (int)(gtid >> 5);
    int  lane = threadIdx.x & 31;
    if (edge >= e) return;

    int r = row[edge];
    int c = col[edge];
    float nrm = dinv[r] * dinv[c];

    float4 v = *(const float4*)(H + (size_t)r * DIM + lane * 4);
    float* o = out + (size_t)c * DIM + lane * 4;
    atomicAdd(o + 0, v.x * nrm);
    atomicAdd(o + 1, v.y * nrm);
    atomicAdd(o + 2, v.z * nrm);
    atomicAdd(o + 3, v.w * nrm);
}

// ---------------- driver ----------------

extern "C" void kernel_launch(void* const* d_in, const int* in_sizes, int n_in,
                              void* d_out, int out_size, void* d_ws, size_t ws_size,
                              hipStream_t stream) {
    const float* x   = (const float*)d_in[0];
    const int*   ei  = (const int*)d_in[1];   // [2, E]: row (src) then col (dst)
    const float* W   = (const float*)d_in[3]; // [3, 128, 128]
    const float* b   = (const float*)d_in[4]; // [3, 128]

    const int n = in_sizes[0] / DIM;          // 100000
    const int e = in_sizes[1] / 2;            // 640000
    const int* rowp = ei;
    const int* colp = ei + e;

    // workspace layout: dinv[n] | hbuf[n*128] | xbuf[n*128]
    float* dinv = (float*)d_ws;
    float* hbuf = dinv + ((n + 255) & ~255);
    float* xbuf = hbuf + (size_t)n * DIM;
    float* deg  = xbuf;                       // reuse xbuf as deg accumulator pre-layer-0

    const long total = (long)n * DIM;
    const int  TPB   = 256;
    const int  gN    = (n + TPB - 1) / TPB;
    const int  gE    = (e + TPB - 1) / TPB;
    const int  gND   = (int)((total + TPB - 1) / TPB);
    const int  gEdge = (int)(((long)e * 32 + TPB - 1) / TPB);
    const int  gTile = (n + 15) / 16;

    // normalization: deg -> dinv
    k_zero<<<gN, TPB, 0, stream>>>(deg, n);
    k_deg_count<<<gE, TPB, 0, stream>>>(colp, deg, e);
    k_dinv<<<gN, TPB, 0, stream>>>(deg, dinv, n);

    const float* xin = x;
    for (int l = 0; l < 3; ++l) {
        const float* Wl = W + (size_t)l * DIM * DIM;
        const float* bl = b + (size_t)l * DIM;
        float* outb = (l == 2) ? (float*)d_out : xbuf;

        k_gemm_wmma<<<gTile, TPB, 0, stream>>>(xin, Wl, hbuf, n);
        k_bias_init<<<gND, TPB, 0, stream>>>(outb, bl, total);
        k_scatter<<<gEdge, TPB, 0, stream>>>(hbuf, rowp, colp, dinv, outb, e);
        if (l < 2) k_relu<<<gND, TPB, 0, stream>>>(outb, total);
        xin = xbuf;
    }
}